// MutualInformationLoss_69106023792751
// MI455X (gfx1250) — compile-verified
//
#include <hip/hip_runtime.h>
#include <hip/hip_bf16.h>

#define NB     32
#define NBINS  (NB * NB)     // 1024
#define BATCH  64
#define EPSF   1e-10f

typedef __attribute__((ext_vector_type(2))) float v2f;
typedef __attribute__((ext_vector_type(8))) float v8f;

// ---------------------------------------------------------------------------
// Kernel 0: zero the per-sample histograms (workspace) and the scalar output.
// Must run every launch: the harness poisons ws/out only once, and we
// accumulate with atomics, so we own re-initialization (graph-replay safe).
// ---------------------------------------------------------------------------
__global__ void mi_zero_kernel(unsigned int* __restrict__ hist,
                               float* __restrict__ out) {
    int i = blockIdx.x * blockDim.x + threadIdx.x;
    if (i < BATCH * NBINS) hist[i] = 0u;
    if (i == 0) out[0] = 0.0f;
}

// ---------------------------------------------------------------------------
// Kernel 1: memory-bound 2D histogram. float4 (b128) loads, LDS-privatized
// counts via ds_add_u32 atomics, then global_atomic_add_u32 merge.
// ---------------------------------------------------------------------------
__global__ void mi_hist_kernel(const float* __restrict__ x,
                               const float* __restrict__ y,
                               unsigned int* __restrict__ hist,
                               int S, int chunks, int chunkQuads) {
    __shared__ unsigned int sh[NBINS];
    const int tid = threadIdx.x;
    for (int i = tid; i < NBINS; i += blockDim.x) sh[i] = 0u;
    __syncthreads();

    const int b = blockIdx.x / chunks;
    const int c = blockIdx.x % chunks;
    const int totalQ = S >> 2;                 // float4s per sample
    const int startQ = c * chunkQuads;
    int quads = chunkQuads;
    if (startQ + quads > totalQ) quads = totalQ - startQ;

    const float4* __restrict__ x4 = (const float4*)x;
    const float4* __restrict__ y4 = (const float4*)y;
    const long long base4 = (long long)b * totalQ + startQ;

    for (int i = tid; i < quads; i += blockDim.x) {
        float4 xv = x4[base4 + i];
        float4 yv = y4[base4 + i];
        float xs[4] = {xv.x, xv.y, xv.z, xv.w};
        float ys[4] = {yv.x, yv.y, yv.z, yv.w};
#pragma unroll
        for (int k = 0; k < 4; ++k) {
            float xx = xs[k], yy = ys[k];
            bool valid = (xx >= 0.0f) && (xx <= 1.0f) &&
                         (yy >= 0.0f) && (yy <= 1.0f);
            int bx = (int)(xx * (float)NB);
            bx = bx < 0 ? 0 : (bx > NB - 1 ? NB - 1 : bx);
            int by = (int)(yy * (float)NB);
            by = by < 0 ? 0 : (by > NB - 1 ? NB - 1 : by);
            if (valid) atomicAdd(&sh[bx * NB + by], 1u);   // ds_add_u32
        }
    }
    __syncthreads();

    unsigned int* __restrict__ g = hist + (size_t)b * NBINS;
    for (int i = tid; i < NBINS; i += blockDim.x) {
        unsigned int v = sh[i];
        if (v) atomicAdd(&g[i], v);                        // global_atomic_add_u32
    }
}

// ---------------------------------------------------------------------------
// Kernel 2: MI epilogue. One wave32 per batch sample. Marginals (row/col sums
// of the 32x32 joint matrix) computed with chained V_WMMA_F32_16X16X4_F32
// against a ones operand (full fp32 precision).
// ---------------------------------------------------------------------------
__device__ __forceinline__ float waveReduceSum32(float v) {
#pragma unroll
    for (int off = 16; off > 0; off >>= 1) v += __shfl_xor(v, off, 32);
    return v;
}

__global__ void mi_reduce_kernel(const unsigned int* __restrict__ hist,
                                 float* __restrict__ out) {
    __shared__ float pj[NBINS];
    __shared__ float rowsum[NB];
    __shared__ float colsum[NB];

    const int b    = blockIdx.x;
    const int lane = threadIdx.x;            // 0..31, single wave32
    const unsigned int* __restrict__ h = hist + (size_t)b * NBINS;

    // Lane L owns row L of the 32x32 joint histogram.
    float rowTot = 0.0f;
    float rowv[NB];
#pragma unroll
    for (int j = 0; j < NB; ++j) {
        rowv[j] = (float)h[lane * NB + j];
        rowTot += rowv[j];
    }
    const float total = waveReduceSum32(rowTot);
    const float inv   = 1.0f / total;
#pragma unroll
    for (int j = 0; j < NB; ++j) pj[lane * NB + j] = rowv[j] * inv;
    __syncthreads();

    const bool hi  = lane >= 16;
    const int  l15 = lane & 15;
    v2f ones; ones[0] = 1.0f; ones[1] = 1.0f;

    // ---- Row sums: D = A(16x4 tile of pj) x B(4x16 ones), chained over K ----
    // 16x4 f32 A layout: lanes 0-15 hold K={0,1}, lanes 16-31 hold K={2,3}.
    for (int half = 0; half < 2; ++half) {
        const int r = half * 16 + l15;
        v8f acc = {};
#pragma unroll
        for (int k = 0; k < 8; ++k) {
            const int c0 = 4 * k + (hi ? 2 : 0);
            v2f a;
            a[0] = pj[r * NB + c0];
            a[1] = pj[r * NB + c0 + 1];
            acc = __builtin_amdgcn_wmma_f32_16x16x4_f32(
                false, a, false, ones, (short)0, acc, false, false);
        }
        // D[i][n] = rowsum(i): VGPR v holds M=v (lanes 0-15) / M=8+v (16-31).
        const int vb = half * 16 + (hi ? 8 : 0);
        if (l15 == 0) {
#pragma unroll
            for (int v = 0; v < 8; ++v) rowsum[vb + v] = acc[v];
        }
    }

    // ---- Col sums: D = A(4x16... ones as 16x4) x B(4x16 tile of pj) ----
    // 4x16 f32 B layout: lanes 0-15 hold K={0,1}, lanes 16-31 hold K={2,3}.
    for (int half = 0; half < 2; ++half) {
        const int cb = half * 16;
        v8f acc = {};
#pragma unroll
        for (int k = 0; k < 8; ++k) {
            const int r0 = 4 * k + (hi ? 2 : 0);
            v2f bb;
            bb[0] = pj[r0 * NB + cb + l15];
            bb[1] = pj[(r0 + 1) * NB + cb + l15];
            acc = __builtin_amdgcn_wmma_f32_16x16x4_f32(
                false, ones, false, bb, (short)0, acc, false, false);
        }
        // D[i][n] = colsum(cb+n), constant over i: VGPR0, lane n gives col cb+n.
        if (!hi) colsum[cb + l15] = acc[0];
    }
    __syncthreads();

    // ---- MI: lane L accumulates row L ----
    float accMI = 0.0f;
    const float px = rowsum[lane] + EPSF;
#pragma unroll
    for (int j = 0; j < NB; ++j) {
        const float p  = pj[lane * NB + j] + EPSF;
        const float py = colsum[j] + EPSF;
        accMI += p * logf(p / (px * py));
    }
    const float mi = waveReduceSum32(accMI);
    if (lane == 0) atomicAdd(out, -mi * (1.0f / (float)BATCH));
}

// ---------------------------------------------------------------------------
extern "C" void kernel_launch(void* const* d_in, const int* in_sizes, int n_in,
                              void* d_out, int out_size, void* d_ws, size_t ws_size,
                              hipStream_t stream) {
    const float* x = (const float*)d_in[0];
    const float* y = (const float*)d_in[1];
    float* out = (float*)d_out;
    unsigned int* hist = (unsigned int*)d_ws;   // BATCH * NBINS u32 = 256 KB

    const int total = in_sizes[0];
    const int S = total / BATCH;                // 524288 for the reference

    // 1) zero workspace + output
    const int zn = BATCH * NBINS;
    mi_zero_kernel<<<(zn + 255) / 256, 256, 0, stream>>>(hist, out);

    // 2) histogram: 64 chunks/sample -> 4096 blocks of 256 threads
    const int chunkQuads = 2048;                // 8192 elements per chunk
    const int totalQ = S >> 2;
    const int chunks = (totalQ + chunkQuads - 1) / chunkQuads;
    mi_hist_kernel<<<BATCH * chunks, 256, 0, stream>>>(x, y, hist, S, chunks, chunkQuads);

    // 3) MI epilogue: one wave32 per sample
    mi_reduce_kernel<<<BATCH, 32, 0, stream>>>(hist, out);
}